// SpatialLoss_4002909519988
// MI455X (gfx1250) — compile-verified
//
#include <hip/hip_runtime.h>

#define IMG_W     256
#define IMG_H     256
#define TILE_H    32
#define LOAD_ROWS 34            // TILE_H + 2 halo rows
#define BLOCK     256
#define SEGS_PER_ROW (IMG_W/4)  // 64 x b128 segments per row

typedef __attribute__((ext_vector_type(2))) float v2f;
typedef __attribute__((ext_vector_type(8))) float v8f;

__global__ void sl_zero_kernel(float* out) {
    if (threadIdx.x == 0 && blockIdx.x == 0) out[0] = 0.0f;
}

__global__ __launch_bounds__(BLOCK) void sl_kernel(const float* __restrict__ x,
                                                   float* __restrict__ out,
                                                   float invNorm)
{
    __shared__ __align__(16) float tile[LOAD_ROWS * IMG_W];
    __shared__ float waveSums[BLOCK / 32];

    const int tid         = threadIdx.x;
    const int tilesPerImg = IMG_H / TILE_H;                 // 8
    const int img         = blockIdx.x / tilesPerImg;
    const int tileRow     = (blockIdx.x - img * tilesPerImg) * TILE_H;

    const float* base = x + (size_t)img * (IMG_W * IMG_H);

    // ---- Stage 34x256 f32 tile into LDS via CDNA5 async global->LDS copies.
    // Rows past the image bottom are clamped to row 255 so LDS always holds
    // finite real data; invalid contributions are masked out later.
    for (int u = tid; u < LOAD_ROWS * SEGS_PER_ROW; u += BLOCK) {
        int row  = u >> 6;           // u / SEGS_PER_ROW
        int seg  = u & 63;
        int grow = tileRow + row;
        grow = (grow > IMG_H - 1) ? (IMG_H - 1) : grow;
        unsigned gofs = (unsigned)((grow * IMG_W + seg * 4) * sizeof(float));
        unsigned lofs = (unsigned)(size_t)(tile + row * IMG_W + seg * 4); // low 32b of flat = LDS addr
        asm volatile("global_load_async_to_lds_b128 %0, %1, %2"
                     :: "v"(lofs), "v"(gofs), "s"(base)
                     : "memory");
    }
    asm volatile("s_wait_asynccnt 0" ::: "memory");
    __syncthreads();

    // ---- Per-thread column; clamped neighbor columns (masked at the end).
    const int c   = tid;
    const int cm2 = (c - 2 < 0) ? 0 : c - 2;
    const int cm1 = (c - 1 < 0) ? 0 : c - 1;
    const int cp1 = (c + 1 > IMG_W - 1) ? IMG_W - 1 : c + 1;
    const int cp2 = (c + 2 > IMG_W - 1) ? IMG_W - 1 : c + 2;

    float a01 = 0.f, a02 = 0.f;
    float a10 = 0.f, a1p1 = 0.f, a1m1 = 0.f, a1p2 = 0.f, a1m2 = 0.f;
    float a20 = 0.f, a2p1 = 0.f, a2m1 = 0.f, a2p2 = 0.f, a2m2 = 0.f;

    // Rows 0..29: always fully valid (tileRow <= 224 => gr+2 <= 255).
    #pragma unroll 6
    for (int r = 0; r < TILE_H - 2; ++r) {
        const float* r0 = tile + r * IMG_W;
        const float* r1 = r0 + IMG_W;
        const float* r2 = r1 + IMG_W;
        float cen = r0[c];
        float d;
        d = r0[cp1] - cen; a01  = fmaf(d, d, a01);
        d = r0[cp2] - cen; a02  = fmaf(d, d, a02);
        d = r1[cm2] - cen; a1m2 = fmaf(d, d, a1m2);
        d = r1[cm1] - cen; a1m1 = fmaf(d, d, a1m1);
        d = r1[c  ] - cen; a10  = fmaf(d, d, a10);
        d = r1[cp1] - cen; a1p1 = fmaf(d, d, a1p1);
        d = r1[cp2] - cen; a1p2 = fmaf(d, d, a1p2);
        d = r2[cm2] - cen; a2m2 = fmaf(d, d, a2m2);
        d = r2[cm1] - cen; a2m1 = fmaf(d, d, a2m1);
        d = r2[c  ] - cen; a20  = fmaf(d, d, a20);
        d = r2[cp1] - cen; a2p1 = fmaf(d, d, a2p1);
        d = r2[cp2] - cen; a2p2 = fmaf(d, d, a2p2);
    }

    // Rows 30..31: row validity masks (only the last tile of an image clips).
    for (int r = TILE_H - 2; r < TILE_H; ++r) {
        const float* r0 = tile + r * IMG_W;
        const float* r1 = r0 + IMG_W;
        const float* r2 = r1 + IMG_W;
        float rv1 = (tileRow + r + 1 <= IMG_H - 1) ? 1.0f : 0.0f;
        float rv2 = (tileRow + r + 2 <= IMG_H - 1) ? 1.0f : 0.0f;
        float cen = r0[c];
        float d;
        d =  r0[cp1] - cen;        a01  = fmaf(d, d, a01);
        d =  r0[cp2] - cen;        a02  = fmaf(d, d, a02);
        d = (r1[cm2] - cen) * rv1; a1m2 = fmaf(d, d, a1m2);
        d = (r1[cm1] - cen) * rv1; a1m1 = fmaf(d, d, a1m1);
        d = (r1[c  ] - cen) * rv1; a10  = fmaf(d, d, a10);
        d = (r1[cp1] - cen) * rv1; a1p1 = fmaf(d, d, a1p1);
        d = (r1[cp2] - cen) * rv1; a1p2 = fmaf(d, d, a1p2);
        d = (r2[cm2] - cen) * rv2; a2m2 = fmaf(d, d, a2m2);
        d = (r2[cm1] - cen) * rv2; a2m1 = fmaf(d, d, a2m1);
        d = (r2[c  ] - cen) * rv2; a20  = fmaf(d, d, a20);
        d = (r2[cp1] - cen) * rv2; a2p1 = fmaf(d, d, a2p1);
        d = (r2[cp2] - cen) * rv2; a2p2 = fmaf(d, d, a2p2);
    }

    // ---- Column-validity masks + symmetric weights (factor 2/sqrt(i^2+j^2)).
    float mp1 = (c + 1 < IMG_W) ? 1.0f : 0.0f;
    float mp2 = (c + 2 < IMG_W) ? 1.0f : 0.0f;
    float mm1 = (c - 1 >= 0)    ? 1.0f : 0.0f;
    float mm2 = (c - 2 >= 0)    ? 1.0f : 0.0f;

    float p = 2.0f                * (a01 * mp1 + a10)
            + 1.0f                * (a02 * mp2 + a20)
            + 1.41421356237310f   * (a1p1 * mp1 + a1m1 * mm1)
            + 0.894427190999916f  * (a1p2 * mp2 + a1m2 * mm2 + a2p1 * mp1 + a2m1 * mm1)
            + 0.707106781186548f  * (a2p2 * mp2 + a2m2 * mm2);

    // ---- Wave reduction via WMMA: A[lane] = (p, 0), B = ones.
    // D[m][n] = p[m] + p[m+16]; summing the 8 D regs gives half-wave sums,
    // xor-shuffle by 16 completes the full wave-32 sum (EXEC all-ones here).
    v2f A; A.x = p;    A.y = 0.0f;
    v2f B; B.x = 1.0f; B.y = 1.0f;
    v8f C = {0.f, 0.f, 0.f, 0.f, 0.f, 0.f, 0.f, 0.f};
    C = __builtin_amdgcn_wmma_f32_16x16x4_f32(false, A, false, B,
                                              (short)0, C, false, false);
    float s = C[0] + C[1] + C[2] + C[3] + C[4] + C[5] + C[6] + C[7];
    s += __shfl_xor(s, 16, 32);

    const int lane = tid & 31;
    const int wv   = tid >> 5;
    if (lane == 0) waveSums[wv] = s;
    __syncthreads();
    if (tid == 0) {
        float bs = 0.0f;
        #pragma unroll
        for (int i = 0; i < BLOCK / 32; ++i) bs += waveSums[i];
        atomicAdd(out, bs * invNorm);  // pre-scaled: accumulator stays small
    }
}

extern "C" void kernel_launch(void* const* d_in, const int* in_sizes, int n_in,
                              void* d_out, int out_size, void* d_ws, size_t ws_size,
                              hipStream_t stream) {
    const float* x   = (const float*)d_in[0];
    float*       out = (float*)d_out;

    const int nImg    = in_sizes[0] / (IMG_W * IMG_H);  // n_times * n_channels = 1024
    const int nTimes  = nImg / 16;                      // n_channels = 16 per reference
    const float invNorm = 1.0f / ((float)nTimes * (float)IMG_W * (float)IMG_H);

    sl_zero_kernel<<<1, 1, 0, stream>>>(out);

    dim3 grid(nImg * (IMG_H / TILE_H));
    sl_kernel<<<grid, BLOCK, 0, stream>>>(x, out, invNorm);
}